// HybridMemory_33414845563631
// MI455X (gfx1250) — compile-verified
//
#include <hip/hip_runtime.h>
#include <hip/hip_bf16.h>
#include <math.h>
#include <stdint.h>

typedef __attribute__((ext_vector_type(2))) float v2f;
typedef __attribute__((ext_vector_type(8))) float v8f;

#define B_N      256
#define D_K      2048
#define P_N      8192
#define N_PP     8
#define N_CAMS   8
#define BG_KNN   40
#define POSK     3
#define INV_SCALE 10.0f

// ---------------------------------------------------------------------------
// Kernel 1: gather A = concat(features, mem[proxy_targets]) -> [512, 2048];
// also compute pt[b] and cams[b].
// ---------------------------------------------------------------------------
__global__ __launch_bounds__(256)
void hm_gather(const float* __restrict__ features,
               const float* __restrict__ mem,
               const int* __restrict__ targets,
               const int* __restrict__ apl,
               float* __restrict__ A,
               int* __restrict__ pt,
               int* __restrict__ cams) {
  const int r = blockIdx.x;          // 0..511
  const int tid = threadIdx.x;       // 0..255
  const float* src;
  if (r < B_N) {
    src = features + (size_t)r * D_K;
  } else {
    const int b = r - B_N;
    const int p = apl[targets[b]];
    src = mem + (size_t)p * D_K;
    if (tid == 0) { pt[b] = p; cams[b] = p % N_CAMS; }
  }
  const float4* s4 = (const float4*)src;
  float4* d4 = (float4*)(A + (size_t)r * D_K);
  d4[tid]       = s4[tid];
  d4[tid + 256] = s4[tid + 256];
}

// ---------------------------------------------------------------------------
// Kernel 2: fused f32 WMMA GEMM  S[512,8192] = A[512,2048] * Mem[8192,2048]^T
// rows 0..255 of S = score, rows 256..511 = sims. Full f32 precision via
// V_WMMA_F32_16X16X4_F32. Block = 256 threads (8 wave32), tile 32x128.
// Double-buffered LDS staging via GLOBAL_LOAD_ASYNC_TO_LDS_B128 (ASYNCcnt),
// overlapping global->LDS traffic with the WMMA pipe.
// ---------------------------------------------------------------------------
#define KC     32
#define PITCH  36   // 16B-aligned rows, conflict-free LDS banking
#define NCHUNK (D_K / KC)

__global__ __launch_bounds__(256)
void hm_gemm_wmma(const float* __restrict__ A,
                  const float* __restrict__ Mem,
                  float* __restrict__ S) {
  __shared__ float Asm[2][32 * PITCH];
  __shared__ float Bsm[2][128 * PITCH];

  const int tid    = threadIdx.x;
  const int wave   = tid >> 5;
  const int lane   = tid & 31;
  const int lane16 = lane & 15;
  const int half   = lane >> 4;
  const int mw     = wave & 1;   // 0..1 -> M sub-tile
  const int nw     = wave >> 1;  // 0..3 -> N sub-tile
  const int m0     = blockIdx.y * 32;
  const int n0     = blockIdx.x * 128;

  const int lr = tid >> 3;        // 0..31
  const int lc = (tid & 7) * 4;   // 0..28

  // Per-lane global source addresses for this thread's staging slice
  const float* gA = A   + (size_t)(m0 + lr) * D_K + lc;   // A row chunk
  const float* gB = Mem + (size_t)(n0 + lr) * D_K + lc;   // B rows lr, lr+32, ...

  v8f acc0 = {};
  v8f acc1 = {};

  // Issue async copies of K-chunk `kc` into LDS buffer `buf`.
  auto issue_chunk = [&](int kc, int buf) {
    unsigned lA = (unsigned)(uintptr_t)&Asm[buf][lr * PITCH + lc];
    unsigned long long aA = (unsigned long long)(uintptr_t)(gA + kc);
    asm volatile("global_load_async_to_lds_b128 %0, %1, off"
                 :: "v"(lA), "v"(aA) : "memory");
#pragma unroll
    for (int i = 0; i < 4; ++i) {
      unsigned lB = (unsigned)(uintptr_t)&Bsm[buf][(i * 32 + lr) * PITCH + lc];
      unsigned long long aB =
          (unsigned long long)(uintptr_t)(gB + (size_t)i * 32 * D_K + kc);
      asm volatile("global_load_async_to_lds_b128 %0, %1, off"
                   :: "v"(lB), "v"(aB) : "memory");
    }
  };

  // Prologue: stage chunk 0
  issue_chunk(0, 0);
  asm volatile("s_wait_asynccnt 0x0" ::: "memory");
  __syncthreads();

  for (int c = 0; c < NCHUNK; ++c) {
    const int cur = c & 1;
    if (c + 1 < NCHUNK) issue_chunk((c + 1) * KC, cur ^ 1);

    const float* As = &Asm[cur][(mw * 16 + lane16) * PITCH];
    const float* Bs0 = &Bsm[cur][(nw * 32 + lane16) * PITCH];
    const float* Bs1 = &Bsm[cur][(nw * 32 + 16 + lane16) * PITCH];

    // 8 k-steps of V_WMMA_F32_16X16X4_F32; each lane feeds contiguous K-pairs
#pragma unroll
    for (int ks = 0; ks < 8; ++ks) {
      const int k = ks * 4 + 2 * half;
      const v2f a  = *(const v2f*)(As + k);
      const v2f b0 = *(const v2f*)(Bs0 + k);
      const v2f b1 = *(const v2f*)(Bs1 + k);
      acc0 = __builtin_amdgcn_wmma_f32_16x16x4_f32(
          false, a, false, b0, (short)0, acc0, false, false);
      acc1 = __builtin_amdgcn_wmma_f32_16x16x4_f32(
          false, a, false, b1, (short)0, acc1, false, false);
    }

    // Next chunk landed (own wave) + all waves done reading `cur`
    asm volatile("s_wait_asynccnt 0x0" ::: "memory");
    __syncthreads();
  }

  // C/D layout: vgpr r holds D[M = r + 8*half][N = lane16]
#pragma unroll
  for (int r = 0; r < 8; ++r) {
    const int gm = m0 + mw * 16 + r + 8 * half;
    const int gn = n0 + nw * 32 + lane16;
    S[(size_t)gm * P_N + gn]      = acc0[r];
    S[(size_t)gm * P_N + gn + 16] = acc1[r];
  }
}

// ---------------------------------------------------------------------------
// Kernel 3: offline loss per row.
// L_off[b] = logsumexp(8 positives ∪ top-40 of rest of score/0.1) - mean(pos)
// ---------------------------------------------------------------------------
__global__ __launch_bounds__(256)
void hm_loss_off(const float* __restrict__ S,
                 const int* __restrict__ pt,
                 float* __restrict__ Loff) {
  __shared__ float vals[P_N];
  __shared__ float red_v[256];
  __shared__ int   red_i[256];
  __shared__ float sel[BG_KNN + N_PP];   // [0..39]=top bg, [40..47]=positives

  const int b = blockIdx.x;
  const int tid = threadIdx.x;
  const float* row = S + (size_t)b * P_N;
  for (int i = tid; i < P_N; i += 256) vals[i] = row[i] * INV_SCALE;
  __syncthreads();

  const int pb = (pt[b] / N_PP) * N_PP;  // positives = pb..pb+7
  if (tid < N_PP) sel[BG_KNN + tid] = vals[pb + tid];
  __syncthreads();
  if (tid < N_PP) vals[pb + tid] = -1e30f;
  __syncthreads();

  for (int it = 0; it < BG_KNN; ++it) {
    float bm = -1e30f; int bi = 0;
    const int base = tid * 32;
#pragma unroll 8
    for (int j = 0; j < 32; ++j) {
      const float v = vals[base + j];
      if (v > bm) { bm = v; bi = base + j; }
    }
    red_v[tid] = bm; red_i[tid] = bi;
    __syncthreads();
    for (int s = 128; s > 0; s >>= 1) {
      if (tid < s && red_v[tid + s] > red_v[tid]) {
        red_v[tid] = red_v[tid + s]; red_i[tid] = red_i[tid + s];
      }
      __syncthreads();
    }
    if (tid == 0) { sel[it] = red_v[0]; vals[red_i[0]] = -1e30f; }
    __syncthreads();
  }

  if (tid == 0) {
    float m = -1e30f;
    for (int i = 0; i < BG_KNN + N_PP; ++i) m = fmaxf(m, sel[i]);
    float se = 0.0f;
    for (int i = 0; i < BG_KNN + N_PP; ++i) se += expf(sel[i] - m);
    float ps = 0.0f;
    for (int i = 0; i < N_PP; ++i) ps += sel[BG_KNN + i];
    Loff[b] = (m + logf(se)) - ps * (1.0f / N_PP);
  }
}

// ---------------------------------------------------------------------------
// Kernel 4: online loss per row (camera-wise NN enhance).
// ---------------------------------------------------------------------------
__global__ __launch_bounds__(256)
void hm_loss_on(const float* __restrict__ S,
                float* __restrict__ Lon) {
  __shared__ float sv[P_N];
  __shared__ float red_v[256];
  __shared__ int   red_i[256];
  __shared__ float camval[N_CAMS];
  __shared__ int   camproxy[N_CAMS];
  __shared__ int   boost[POSK];
  __shared__ float sel[BG_KNN + POSK];   // [0..2]=boost inputs, [3..42]=bg inputs

  const int b = blockIdx.x;
  const int tid = threadIdx.x;
  const float* srow = S + (size_t)(B_N + b) * P_N;  // sims row
  const float* irow = S + (size_t)b * P_N;          // score row (inputs = *10)
  for (int i = tid; i < P_N; i += 256) sv[i] = srow[i];
  __syncthreads();

  // camera-wise argmax: slot tid = g*8 + c; thread scans ids g*32..g*32+31
  {
    const int c = tid & 7, g = tid >> 3;
    float bm = -1e30f; int bid = 0;
    for (int j = 0; j < 32; ++j) {
      const int id = g * 32 + j;
      const float v = sv[id * N_CAMS + c];
      if (v > bm) { bm = v; bid = id; }
    }
    red_v[tid] = bm; red_i[tid] = bid;
    __syncthreads();
    for (int s = 16; s > 0; s >>= 1) {
      if ((tid >> 3) < s) {
        const int o = tid + s * 8;
        if (red_v[o] > red_v[tid]) { red_v[tid] = red_v[o]; red_i[tid] = red_i[o]; }
      }
      __syncthreads();
    }
    if (tid < N_CAMS) {
      camval[tid] = red_v[tid];
      camproxy[tid] = red_i[tid] * N_CAMS + tid;
    }
  }
  __syncthreads();

  if (tid == 0) {
    unsigned used = 0;
    for (int k = 0; k < POSK; ++k) {
      float bm = -1e38f; int bc = 0;
      for (int c = 0; c < N_CAMS; ++c) {
        if (used & (1u << c)) continue;
        if (camval[c] > bm) { bm = camval[c]; bc = c; }
      }
      used |= 1u << bc;
      boost[k] = camproxy[bc];
    }
    for (int k = 0; k < POSK; ++k) {
      sel[k] = irow[boost[k]] * INV_SCALE;
      sv[boost[k]] = -1e30f;
    }
  }
  __syncthreads();

  for (int it = 0; it < BG_KNN; ++it) {
    float bm = -1e30f; int bi = 0;
    const int base = tid * 32;
#pragma unroll 8
    for (int j = 0; j < 32; ++j) {
      const float v = sv[base + j];
      if (v > bm) { bm = v; bi = base + j; }
    }
    red_v[tid] = bm; red_i[tid] = bi;
    __syncthreads();
    for (int s = 128; s > 0; s >>= 1) {
      if (tid < s && red_v[tid + s] > red_v[tid]) {
        red_v[tid] = red_v[tid + s]; red_i[tid] = red_i[tid + s];
      }
      __syncthreads();
    }
    if (tid == 0) {
      const int idx = red_i[0];
      sel[POSK + it] = irow[idx] * INV_SCALE;  // inputs at sims-selected index
      sv[idx] = -1e30f;
    }
    __syncthreads();
  }

  if (tid == 0) {
    float m = -1e30f;
    for (int i = 0; i < BG_KNN + POSK; ++i) m = fmaxf(m, sel[i]);
    float se = 0.0f;
    for (int i = 0; i < BG_KNN + POSK; ++i) se += expf(sel[i] - m);
    const float bs = sel[0] + sel[1] + sel[2];
    Lon[b] = (m + logf(se)) - bs * (1.0f / POSK);
  }
}

// ---------------------------------------------------------------------------
// Kernel 5: per-camera mean then sum -> scalar loss.
// ---------------------------------------------------------------------------
__global__ __launch_bounds__(32)
void hm_finalize(const float* __restrict__ Loff,
                 const float* __restrict__ Lon,
                 const int* __restrict__ cams,
                 float* __restrict__ out) {
  __shared__ float so[N_CAMS], sn[N_CAMS];
  __shared__ int   sc[N_CAMS];
  const int tid = threadIdx.x;
  if (tid < N_CAMS) {
    float o = 0.0f, n = 0.0f; int c = 0;
    for (int b = 0; b < B_N; ++b) {
      if (cams[b] == tid) { o += Loff[b]; n += Lon[b]; ++c; }
    }
    so[tid] = o; sn[tid] = n; sc[tid] = c;
  }
  __syncthreads();
  if (tid == 0) {
    float loss = 0.0f;
    for (int c = 0; c < N_CAMS; ++c)
      if (sc[c] > 0) loss += (so[c] + sn[c]) / (float)sc[c];
    out[0] = loss;
  }
}

// ---------------------------------------------------------------------------
extern "C" void kernel_launch(void* const* d_in, const int* in_sizes, int n_in,
                              void* d_out, int out_size, void* d_ws, size_t ws_size,
                              hipStream_t stream) {
  const float* features = (const float*)d_in[0];   // [256, 2048]
  const float* mem      = (const float*)d_in[1];   // [8192, 2048]
  const int*   targets  = (const int*)d_in[2];     // [256]
  const int*   apl      = (const int*)d_in[3];     // [16384]
  float* out = (float*)d_out;

  // workspace layout (~21 MB)
  char* ws = (char*)d_ws;
  float* A    = (float*)ws;                                   // 512*2048 f32 (4 MB)
  float* S    = (float*)(ws + (size_t)512 * D_K * 4);         // 512*8192 f32 (16 MB)
  char*  tail = ws + (size_t)512 * D_K * 4 + (size_t)512 * P_N * 4;
  float* Loff = (float*)tail;                                 // 256 f32
  float* Lon  = (float*)(tail + 1024);                        // 256 f32
  int*   pt   = (int*)(tail + 2048);                          // 256 i32
  int*   cams = (int*)(tail + 3072);                          // 256 i32

  hm_gather<<<512, 256, 0, stream>>>(features, mem, targets, apl, A, pt, cams);
  hm_gemm_wmma<<<dim3(P_N / 128, 512 / 32), 256, 0, stream>>>(A, mem, S);
  hm_loss_off<<<B_N, 256, 0, stream>>>(S, pt, Loff);
  hm_loss_on<<<B_N, 256, 0, stream>>>(S, Lon);
  hm_finalize<<<1, 32, 0, stream>>>(Loff, Lon, cams, out);
}